// DenseGrid_52029233823908
// MI455X (gfx1250) — compile-verified
//
#include <hip/hip_runtime.h>

// Multi-LOD dense-grid trilinear interpolation for MI455X (gfx1250, wave32).
// Gather-bound kernel: 8x global_load_b128 per LOD per point (FEAT_DIM=4 f32
// == one 16B vector), NT hints on streaming pts/out to keep the 192MB L2 for
// grid data, grid3 (256MB, HBM) gathers issued first for latency overlap.

typedef __attribute__((ext_vector_type(4))) float v4f;

template <int LOD>
__device__ __forceinline__ v4f trilerp_lod(const v4f* __restrict__ G,
                                           float px, float py, float pz) {
  constexpr float MX = (float)(LOD - 1);
  // c = clip(((p + 1) * 0.5) * (LOD-1), 0, LOD-1)  -- matches reference assoc.
  float cx = fminf(fmaxf((px + 1.0f) * 0.5f * MX, 0.0f), MX);
  float cy = fminf(fmaxf((py + 1.0f) * 0.5f * MX, 0.0f), MX);
  float cz = fminf(fmaxf((pz + 1.0f) * 0.5f * MX, 0.0f), MX);

  float fx0 = floorf(cx), fy0 = floorf(cy), fz0 = floorf(cz);
  float fx = cx - fx0, fy = cy - fy0, fz = cz - fz0;
  float gx = 1.0f - fx, gy = 1.0f - fy, gz = 1.0f - fz;

  int x0 = (int)fx0, y0 = (int)fy0, z0 = (int)fz0;
  int x1 = min(x0 + 1, LOD - 1);
  int y1 = min(y0 + 1, LOD - 1);
  int z1 = min(z0 + 1, LOD - 1);

  // flat(x,y,z) = (x*LOD + y)*LOD + z ; elements are v4f (16B) -> b128 gathers
  int r00 = (x0 * LOD + y0) * LOD;
  int r01 = (x0 * LOD + y1) * LOD;
  int r10 = (x1 * LOD + y0) * LOD;
  int r11 = (x1 * LOD + y1) * LOD;

  // Issue all 8 gathers before any consumption (MLP within the thread).
  v4f f000 = G[r00 + z0];
  v4f f001 = G[r00 + z1];
  v4f f010 = G[r01 + z0];
  v4f f011 = G[r01 + z1];
  v4f f100 = G[r10 + z0];
  v4f f101 = G[r10 + z1];
  v4f f110 = G[r11 + z0];
  v4f f111 = G[r11 + z1];

  float w000 = gx * gy * gz, w001 = gx * gy * fz;
  float w010 = gx * fy * gz, w011 = gx * fy * fz;
  float w100 = fx * gy * gz, w101 = fx * gy * fz;
  float w110 = fx * fy * gz, w111 = fx * fy * fz;

  // Same accumulation order as reference (000 -> 111); -ffp-contract => v_fma.
  v4f acc = f000 * w000;
  acc += f001 * w001;
  acc += f010 * w010;
  acc += f011 * w011;
  acc += f100 * w100;
  acc += f101 * w101;
  acc += f110 * w110;
  acc += f111 * w111;
  return acc;
}

__global__ __launch_bounds__(256) void densegrid_trilerp_kernel(
    const float* __restrict__ pts,
    const v4f* __restrict__ g0, const v4f* __restrict__ g1,
    const v4f* __restrict__ g2, const v4f* __restrict__ g3,
    v4f* __restrict__ out, int n) {
  int i = blockIdx.x * blockDim.x + threadIdx.x;
  if (i >= n) return;

  // Streaming, single-touch: non-temporal loads keep pts out of L2.
  size_t p = (size_t)i * 3;
  float px = __builtin_nontemporal_load(pts + p + 0);
  float py = __builtin_nontemporal_load(pts + p + 1);
  float pz = __builtin_nontemporal_load(pts + p + 2);

  // Biggest grid first: its HBM-miss latency overlaps the L2-resident LODs.
  v4f r3 = trilerp_lod<256>(g3, px, py, pz);
  v4f r2 = trilerp_lod<128>(g2, px, py, pz);
  v4f r1 = trilerp_lod<64>(g1, px, py, pz);
  v4f r0 = trilerp_lod<32>(g0, px, py, pz);

  // Output row = 16 floats = 4x b128 NT stores; a wave's 4 store instructions
  // jointly cover a contiguous 2KB region, and NT keeps the 128MB output
  // stream from evicting grid lines in L2.
  v4f* o = out + (size_t)i * 4;
  __builtin_nontemporal_store(r0, o + 0);
  __builtin_nontemporal_store(r1, o + 1);
  __builtin_nontemporal_store(r2, o + 2);
  __builtin_nontemporal_store(r3, o + 3);
}

extern "C" void kernel_launch(void* const* d_in, const int* in_sizes, int n_in,
                              void* d_out, int out_size, void* d_ws, size_t ws_size,
                              hipStream_t stream) {
  const float* pts = (const float*)d_in[0];
  const v4f* g0 = (const v4f*)d_in[1];
  const v4f* g1 = (const v4f*)d_in[2];
  const v4f* g2 = (const v4f*)d_in[3];
  const v4f* g3 = (const v4f*)d_in[4];
  v4f* out = (v4f*)d_out;

  int n = in_sizes[0] / 3;  // 2,000,000 points
  const int block = 256;    // 8 wave32 waves per block
  int grid = (n + block - 1) / block;
  hipLaunchKernelGGL(densegrid_trilerp_kernel, dim3(grid), dim3(block), 0,
                     stream, pts, g0, g1, g2, g3, out, n);
}